// MKLCSRSparseMatrixGCN_80247168959056
// MI455X (gfx1250) — compile-verified
//
#include <hip/hip_runtime.h>
#include <hip/hip_bf16.h>

typedef __attribute__((ext_vector_type(2))) float v2f;
typedef __attribute__((ext_vector_type(8))) float v8f;

#define D_FEAT 64

// ---------------------------------------------------------------------------
// 1) zero out[] (N*64) and deg[] (N)
// ---------------------------------------------------------------------------
__global__ void gcn_zero_kernel(float* __restrict__ out, float* __restrict__ deg,
                                int nout, int ndeg) {
    int i = blockIdx.x * blockDim.x + threadIdx.x;
    if (i < nout) out[i] = 0.0f;
    if (i < ndeg) deg[i] = 0.0f;
}

// ---------------------------------------------------------------------------
// 2) degree: deg[row[e]] += 1  (float accumulation, exact for deg < 2^24)
// ---------------------------------------------------------------------------
__global__ void gcn_degree_kernel(const int* __restrict__ rows,
                                  float* __restrict__ deg, int nedges) {
    int e = blockIdx.x * blockDim.x + threadIdx.x;
    if (e < nedges) {
        atomicAdd(&deg[rows[e]], 1.0f);
    }
}

// ---------------------------------------------------------------------------
// 3) dinv_sqrt[i] = deg[i] > 0 ? 1/sqrt(deg[i]) : 0
// ---------------------------------------------------------------------------
__global__ void gcn_dinv_kernel(const float* __restrict__ deg,
                                float* __restrict__ dinv, int n) {
    int i = blockIdx.x * blockDim.x + threadIdx.x;
    if (i < n) {
        float d = deg[i];
        dinv[i] = (d > 0.0f) ? (1.0f / sqrtf(d)) : 0.0f;
    }
}

// ---------------------------------------------------------------------------
// 4) xh = x @ W^T via V_WMMA_F32_16X16X4_F32.
//    One wave32 computes one 16x16 tile of xh. K loop: 64 / 4 = 16 WMMA steps.
//
//    Fragment layouts (CDNA5 ISA 7.12.2, 32-bit data):
//      A (16x4, MxK):  lanes 0-15 M=lane, {V0,V1}={K0,K1};
//                      lanes 16-31 M=lane-16, {V0,V1}={K2,K3}
//      B (4x16, KxN):  lanes 0-15 N=lane, {V0,V1}={K0,K1};
//                      lanes 16-31 N=lane-16, {V0,V1}={K2,K3}
//      C/D (16x16):    VGPR r -> M=r (lanes 0-15) / M=r+8 (lanes 16-31), N=lane%16
//    B here is W^T, i.e. B[k][n] = W[n][k], so both A and B fragments are
//    contiguous 2-float (8B) loads along the K axis.
// ---------------------------------------------------------------------------
__global__ void gcn_gemm_wmma_kernel(const float* __restrict__ x,
                                     const float* __restrict__ W,
                                     float* __restrict__ xh, int nrows) {
    const int wave = threadIdx.x >> 5;
    const int lane = threadIdx.x & 31;
    const int waves_per_blk = blockDim.x >> 5;
    const int ntiles_n = D_FEAT / 16;                     // 4
    const int tile = blockIdx.x * waves_per_blk + wave;   // wave-uniform
    const int ntiles_m = (nrows + 15) >> 4;
    if (tile >= ntiles_m * ntiles_n) return;              // whole-wave exit

    const int mtile = tile / ntiles_n;
    const int ntile = tile % ntiles_n;

    const int half  = lane >> 4;          // 0: lanes 0-15, 1: lanes 16-31
    const int l16   = lane & 15;
    const int khalf = half * 2;           // K offset {0,2} within each K=4 step

    int m = mtile * 16 + l16;
    if (m >= nrows) m = nrows - 1;        // clamp loads for ragged tail
    const int n = ntile * 16 + l16;

    const float* xrow = x + (size_t)m * D_FEAT;
    const float* wrow = W + (size_t)n * D_FEAT;           // row n of W == col n of W^T

    v8f c = {};
#pragma unroll
    for (int k0 = 0; k0 < D_FEAT; k0 += 4) {
        v2f a = *(const v2f*)(xrow + k0 + khalf);         // 8B aligned
        v2f b = *(const v2f*)(wrow + k0 + khalf);
        // 8 args: (neg_a, A, neg_b, B, c_mod, C, reuse_a, reuse_b)
        c = __builtin_amdgcn_wmma_f32_16x16x4_f32(
                false, a, false, b, (short)0, c, false, false);
    }

    // D layout: VGPR r holds row (mtile*16 + r + 8*half), col = ntile*16 + l16
    const int row0 = mtile * 16 + half * 8;
    const int colg = ntile * 16 + l16;
#pragma unroll
    for (int r = 0; r < 8; ++r) {
        const int row = row0 + r;
        if (row < nrows) {
            xh[(size_t)row * D_FEAT + colg] = c[r];
        }
    }
}

// ---------------------------------------------------------------------------
// 5) SpMM scatter: out[row] += dinv[row]*dinv[col] * xh[col]
//    16 threads per edge, each owns a float4 (16B) slice -> coalesced 128-bit
//    gathers from L2-resident xh, 4x global_atomic_add_f32 into out.
// ---------------------------------------------------------------------------
__global__ void gcn_spmm_kernel(const int* __restrict__ rows,
                                const int* __restrict__ cols,
                                const float* __restrict__ dinv,
                                const float* __restrict__ xh,
                                float* __restrict__ out, int nedges) {
    const long long tid = (long long)blockIdx.x * blockDim.x + threadIdx.x;
    const int e = (int)(tid >> 4);
    const int t = (int)(tid & 15);
    if (e >= nedges) return;

    const int r = rows[e];
    const int cidx = cols[e];
    const float w = dinv[r] * dinv[cidx];

    const float4 v = *(const float4*)(xh + (size_t)cidx * D_FEAT + t * 4);
    float* op = out + (size_t)r * D_FEAT + t * 4;
    atomicAdd(op + 0, w * v.x);
    atomicAdd(op + 1, w * v.y);
    atomicAdd(op + 2, w * v.z);
    atomicAdd(op + 3, w * v.w);
}

// ---------------------------------------------------------------------------
// launch
// ---------------------------------------------------------------------------
extern "C" void kernel_launch(void* const* d_in, const int* in_sizes, int n_in,
                              void* d_out, int out_size, void* d_ws, size_t ws_size,
                              hipStream_t stream) {
    const int* eidx = (const int*)d_in[0];     // [2, E] int32
    const float* x  = (const float*)d_in[1];   // [N, 64]
    const float* W  = (const float*)d_in[2];   // [64, 64]
    float* out      = (float*)d_out;           // [N, 64]

    const int E = in_sizes[0] / 2;
    const int N = in_sizes[1] / D_FEAT;

    const int* rows = eidx;
    const int* cols = eidx + E;

    // workspace: deg[N] | dinv[N] | xh[N*64]
    float* deg  = (float*)d_ws;
    float* dinv = deg + N;
    float* xh   = dinv + N;

    // 1) zero out + deg
    {
        const int total = N * D_FEAT;
        const int blocks = (total + 255) / 256;
        gcn_zero_kernel<<<blocks, 256, 0, stream>>>(out, deg, total, N);
    }
    // 2) degrees
    {
        const int blocks = (E + 255) / 256;
        gcn_degree_kernel<<<blocks, 256, 0, stream>>>(rows, deg, E);
    }
    // 3) dinv_sqrt
    {
        const int blocks = (N + 255) / 256;
        gcn_dinv_kernel<<<blocks, 256, 0, stream>>>(deg, dinv, N);
    }
    // 4) xh = x @ W^T  (WMMA f32)
    {
        const int ntiles = ((N + 15) / 16) * (D_FEAT / 16);
        const int waves_per_blk = 8;                     // 256 threads
        const int blocks = (ntiles + waves_per_blk - 1) / waves_per_blk;
        gcn_gemm_wmma_kernel<<<blocks, 256, 0, stream>>>(x, W, xh, N);
    }
    // 5) SpMM scatter
    {
        const long long threads = (long long)E * 16;
        const int blocks = (int)((threads + 255) / 256);
        gcn_spmm_kernel<<<blocks, 256, 0, stream>>>(rows, cols, dinv, xh, out, E);
    }
}